// GaussianProcess_42786464202883
// MI455X (gfx1250) — compile-verified
//
#include <hip/hip_runtime.h>
#include <math.h>

typedef __attribute__((ext_vector_type(2))) float v2f;
typedef __attribute__((ext_vector_type(4))) float v4f;
typedef __attribute__((ext_vector_type(8))) float v8f;

#define B_ 8
#define T_ 1024
#define D_ 16
#define N_ (B_ * T_)
#define EPS_T 1e-6f

// ---------------------------------------------------------------------------
// K1: per-column median of variances (N=8192 values per column, 16 columns).
// One workgroup per column; bitonic sort of 8192 floats in LDS (32 KB).
// thr[d] = 0.5*(sorted[4095] + sorted[4096])   (jnp.quantile linear, q=0.5)
// ---------------------------------------------------------------------------
__global__ void gp_median_kernel(const float* __restrict__ vf,
                                 float* __restrict__ thr) {
    __shared__ float s[N_];
    const int d = blockIdx.x;
    const int tid = threadIdx.x; // 1024 threads

    for (int e = 0; e < 8; ++e)
        s[tid + e * 1024] = vf[(size_t)(tid + e * 1024) * D_ + d];
    __syncthreads();

    for (int k = 2; k <= N_; k <<= 1) {
        for (int j = k >> 1; j > 0; j >>= 1) {
            for (int e = 0; e < 8; ++e) {
                int i = tid + e * 1024;
                int ixj = i ^ j;
                if (ixj > i) {
                    bool up = ((i & k) == 0);
                    float a = s[i], b = s[ixj];
                    if ((a > b) == up) { s[i] = b; s[ixj] = a; }
                }
            }
            __syncthreads();
        }
    }
    if (tid == 0) thr[d] = 0.5f * (s[N_ / 2 - 1] + s[N_ / 2]);
}

// ---------------------------------------------------------------------------
// K2: batch-reduce m and V=m*z over B. Xs[tau][0..15]=sum_b m, [16..31]=sum_b V
// (This collapses the 8192-row matmul operand to 1024 rows, since
//  W[i,j] depends only on i%T, j%T.)
// ---------------------------------------------------------------------------
__global__ void gp_colsum_kernel(const float* __restrict__ z,
                                 const float* __restrict__ v,
                                 const float* __restrict__ thr,
                                 float* __restrict__ Xs) {
    int gid = blockIdx.x * 256 + threadIdx.x; // 0..16383
    int tau = gid >> 4;
    int d = gid & 15;
    float th = thr[d];
    float sm = 0.f, sv = 0.f;
    for (int b = 0; b < B_; ++b) {
        size_t idx = ((size_t)(b * T_ + tau)) * D_ + d;
        float m = (v[idx] <= th) ? 1.f : 0.f;
        sm += m;
        sv += m * z[idx];
    }
    Xs[tau * 32 + d] = sm;
    Xs[tau * 32 + 16 + d] = sv;
}

// ---------------------------------------------------------------------------
// K3: WMT[t][c] = sum_tau (1/(|t-tau|+eps)) * Xs[tau][c],  t in [0,1024), c in [0,32)
// via V_WMMA_F32_16X16X4_F32. 64 M-tiles x 2 N-tiles = 128 waves, each wave
// runs 256 k-steps of 16x16x4. A-matrix (Toeplitz W) is generated in VALU with
// v_rcp_f32 (TRANS op -> co-executes with the WMMA/XDL pipe), zero A traffic.
// A layout (ISA 7.12.2, 32-bit A 16x4): lanes 0-15 hold K=0(V0),K=1(V1);
// lanes 16-31 hold K=2(V0),K=3(V1). B 4x16 mirrors it; D: VGPR r = rows r, r+8.
// ---------------------------------------------------------------------------
__global__ void gp_wmma_kernel(const float* __restrict__ Xs,
                               float* __restrict__ WMt) {
    const int wave = (blockIdx.x * blockDim.x + threadIdx.x) >> 5; // 0..127
    const int lane = threadIdx.x & 31;
    const int tile_m = wave >> 1;        // 0..63
    const int c0 = (wave & 1) * 16;      // 0 or 16
    const int row = lane & 15;           // M row (for A) / N col (for B,D)
    const int half = lane >> 4;          // 0: K=0,1 ; 1: K=2,3
    const int t = tile_m * 16 + row;

    v8f acc = {};
    for (int kb = 0; kb < T_; kb += 4) {
        const int tau = kb + half * 2;
        v2f a, b;
        // v_rcp_f32 instead of IEEE divide: single TRANS op, hides under WMMA
        a.x = __builtin_amdgcn_rcpf(fabsf((float)(t - tau)) + EPS_T);
        a.y = __builtin_amdgcn_rcpf(fabsf((float)(t - tau - 1)) + EPS_T);
        b.x = Xs[(tau + 0) * 32 + c0 + row];
        b.y = Xs[(tau + 1) * 32 + c0 + row];
        acc = __builtin_amdgcn_wmma_f32_16x16x4_f32(
            /*neg_a=*/false, a, /*neg_b=*/false, b,
            /*c_mod=*/(short)0, acc, /*reuse_a=*/false, /*reuse_b=*/false);
    }
    // D-matrix 16x16 f32 layout: VGPR r -> M=r (lanes 0-15), M=r+8 (lanes 16-31)
    for (int r = 0; r < 8; ++r)
        WMt[(tile_m * 16 + half * 8 + r) * 32 + c0 + row] = acc[r];
}

// ---------------------------------------------------------------------------
// K4: per-column reductions -> length_scale -> ls2inv[d] = 1/ls^2
// S = 2*(sum U*Wm - sum V*WV); ls = n>=2 ? sqrt(S/n^2) : exp(length_scales[d])
// ---------------------------------------------------------------------------
__device__ __forceinline__ float gp_block_reduce(float v, float* red) {
    int tid = threadIdx.x;
    red[tid] = v;
    __syncthreads();
    for (int s = 128; s > 0; s >>= 1) {
        if (tid < s) red[tid] += red[tid + s];
        __syncthreads();
    }
    float r = red[0];
    __syncthreads();
    return r;
}

__global__ void gp_stats_kernel(const float* __restrict__ z,
                                const float* __restrict__ v,
                                const float* __restrict__ thr,
                                const float* __restrict__ WMt,
                                const float* __restrict__ ls_in,
                                float* __restrict__ ls2inv) {
    __shared__ float red[256];
    const int d = blockIdx.x;
    const int tid = threadIdx.x;
    const float th = thr[d];
    float n = 0.f, s1 = 0.f, s2 = 0.f;
    for (int i = tid; i < N_; i += 256) {
        float zz = z[(size_t)i * D_ + d];
        float vv = v[(size_t)i * D_ + d];
        float m = (vv <= th) ? 1.f : 0.f;
        float V = m * zz;
        float U = V * zz;
        int t = i & (T_ - 1);
        s1 += U * WMt[t * 32 + d];
        s2 += V * WMt[t * 32 + 16 + d];
        n += m;
    }
    n = gp_block_reduce(n, red);
    s1 = gp_block_reduce(s1, red);
    s2 = gp_block_reduce(s2, red);
    if (tid == 0) {
        float S = 2.f * (s1 - s2);
        float ls = (n >= 2.f) ? sqrtf(S / (n * n)) : expf(ls_in[d]);
        ls2inv[d] = 1.0f / (ls * ls);
    }
}

// ---------------------------------------------------------------------------
// K5: table[d][delta] = exp(-delta^2 / ls[d]^2) — only D*T=16384 unique exps
// exist in the entire 512 MB output.
// ---------------------------------------------------------------------------
__global__ void gp_table_kernel(const float* __restrict__ ls2inv,
                                float* __restrict__ table) {
    int d = blockIdx.x;
    int delta = threadIdx.x; // 1024
    float x = (float)(delta * delta);
    table[d * T_ + delta] = expf(-x * ls2inv[d]);
}

// ---------------------------------------------------------------------------
// K6: out[b,d,t,t'] = table[d][|t-t'|], 8 identical b-copies. Pure streaming
// store kernel: table row cached in LDS, each thread emits 8 x b128
// NON-TEMPORAL stores (TH=NT) — 512 MB written once, never re-read; don't
// cycle it through L2. 512 MB / 23.3 TB/s ~= 22 us floor; this kernel IS
// the runtime.
// ---------------------------------------------------------------------------
__global__ void gp_fill_kernel(const float* __restrict__ table,
                               float* __restrict__ out) {
    __shared__ float row[T_];
    const int t = blockIdx.x;
    const int d = blockIdx.y;
    const int tid = threadIdx.x; // 256
    for (int e = 0; e < 4; ++e)
        row[tid + e * 256] = table[d * T_ + tid + e * 256];
    __syncthreads();

    const int tp = tid * 4;
    v4f val;
    val.x = row[abs(t - tp)];
    val.y = row[abs(t - (tp + 1))];
    val.z = row[abs(t - (tp + 2))];
    val.w = row[abs(t - (tp + 3))];

    #pragma unroll
    for (int b = 0; b < B_; ++b) {
        size_t off = (((size_t)(b * D_ + d)) * T_ + t) * T_ + tp;
        __builtin_nontemporal_store(val, (v4f*)(out + off));
    }
}

// ---------------------------------------------------------------------------
extern "C" void kernel_launch(void* const* d_in, const int* in_sizes, int n_in,
                              void* d_out, int out_size, void* d_ws, size_t ws_size,
                              hipStream_t stream) {
    const float* z = (const float*)d_in[0];       // (B,T,D)
    const float* vf = (const float*)d_in[1];      // (B,T,D)
    const float* ls_in = (const float*)d_in[2];   // (D,)
    // d_in[3] = sigmas: unused by the output (K depends only on length_scale)
    float* out = (float*)d_out;                   // (B,D,T,T) f32

    char* ws = (char*)d_ws;
    float* thr    = (float*)(ws);                       // 16 f
    float* ls2inv = (float*)(ws + 256);                 // 16 f
    float* Xs     = (float*)(ws + 4096);                // 1024*32 f = 128 KB
    float* WMt    = (float*)(ws + 4096 + 131072);       // 1024*32 f = 128 KB
    float* table  = (float*)(ws + 4096 + 262144);       // 16*1024 f = 64 KB

    gp_median_kernel<<<D_, 1024, 0, stream>>>(vf, thr);
    gp_colsum_kernel<<<(T_ * D_) / 256, 256, 0, stream>>>(z, vf, thr, Xs);
    gp_wmma_kernel<<<32, 128, 0, stream>>>(Xs, WMt);   // 128 waves, 32768 WMMAs
    gp_stats_kernel<<<D_, 256, 0, stream>>>(z, vf, thr, WMt, ls_in, ls2inv);
    gp_table_kernel<<<D_, T_, 0, stream>>>(ls2inv, table);
    gp_fill_kernel<<<dim3(T_, D_, 1), 256, 0, stream>>>(table, out);
}